// HWnet_base_9096740733130
// MI455X (gfx1250) — compile-verified
//
#include <hip/hip_runtime.h>
#include <stdint.h>

// HWnet window-softmax lookup for MI455X (gfx1250, wave32).
//
// Per-sample work: binary search over a sorted 1024-entry table held in LDS,
// then a 17-tap takecare-sharpened softmax over vector_table.
// Tables are staged into LDS with the CDNA5 async global->LDS copy path
// (GLOBAL_LOAD_ASYNC_TO_LDS_B128/B32, tracked by ASYNCcnt), and the
// per-thread input load is issued underneath the DMA to hide its latency.

#define EDGE      8
#define WIN       (2 * EDGE + 1)
#define BLOCK     256
#define TABLE_MAX 1024

__global__ __launch_bounds__(BLOCK)
void hwnet_window_softmax(const float* __restrict__ inputs,
                          const float* __restrict__ eval_t,
                          const float* __restrict__ tk_t,
                          const float* __restrict__ vec_t,
                          float* __restrict__ out,
                          int n, int tlen)
{
    __shared__ float s_eval[TABLE_MAX];
    __shared__ float s_tk[TABLE_MAX];
    __shared__ float s_vec[TABLE_MAX];

    const int tid = threadIdx.x;

    // ---- Stage tables into LDS via CDNA5 async copies (ASYNCcnt path) ----
    // Wide path: 16 bytes per lane per table -> 1024 floats in one pass.
    const int nv4 = tlen >> 2;            // number of float4 chunks
    for (int i = tid; i < nv4; i += BLOCK) {
        const int e = i << 2;
        uint32_t lds_e = (uint32_t)(uintptr_t)(&s_eval[e]);
        uint32_t lds_t = (uint32_t)(uintptr_t)(&s_tk[e]);
        uint32_t lds_v = (uint32_t)(uintptr_t)(&s_vec[e]);
        asm volatile("global_load_async_to_lds_b128 %0, %1, off"
                     :: "v"(lds_e), "v"(eval_t + e) : "memory");
        asm volatile("global_load_async_to_lds_b128 %0, %1, off"
                     :: "v"(lds_t), "v"(tk_t + e) : "memory");
        asm volatile("global_load_async_to_lds_b128 %0, %1, off"
                     :: "v"(lds_v), "v"(vec_t + e) : "memory");
    }
    // Tail (tlen not a multiple of 4): scalar async copies.
    for (int i = (nv4 << 2) + tid; i < tlen; i += BLOCK) {
        uint32_t lds_e = (uint32_t)(uintptr_t)(&s_eval[i]);
        uint32_t lds_t = (uint32_t)(uintptr_t)(&s_tk[i]);
        uint32_t lds_v = (uint32_t)(uintptr_t)(&s_vec[i]);
        asm volatile("global_load_async_to_lds_b32 %0, %1, off"
                     :: "v"(lds_e), "v"(eval_t + i) : "memory");
        asm volatile("global_load_async_to_lds_b32 %0, %1, off"
                     :: "v"(lds_t), "v"(tk_t + i) : "memory");
        asm volatile("global_load_async_to_lds_b32 %0, %1, off"
                     :: "v"(lds_v), "v"(vec_t + i) : "memory");
    }

    // ---- Overlap: issue the per-thread input load under the DMA ----
    const int gid = blockIdx.x * BLOCK + tid;
    const bool valid = (gid < n);
    float x = 0.0f;
    if (valid) x = inputs[gid];           // LOADcnt op, overlaps ASYNCcnt DMA

    asm volatile("s_wait_asynccnt 0" ::: "memory");
    __syncthreads();

    if (!valid) return;

    // ---- lower_bound on sorted table: first index with eval >= x ----
    int lo = 0, hi = tlen;
    while (lo < hi) {
        int mid = (lo + hi) >> 1;
        if (s_eval[mid] < x) lo = mid + 1; else hi = mid;
    }

    // Nearest index, argmin tie -> lower index (dl <= dr picks lo-1).
    int idx;
    if (lo <= 0) {
        idx = 0;
    } else if (lo >= tlen) {
        idx = tlen - 1;
    } else {
        float dl = x - s_eval[lo - 1];   // > 0
        float dr = s_eval[lo] - x;       // >= 0
        idx = (dl <= dr) ? (lo - 1) : lo;
    }

    const float tk = s_tk[idx];          // takecare at UNCLIPPED idx (matches ref)
    int c = idx;
    if (c < EDGE) c = EDGE;
    if (c > tlen - EDGE - 1) c = tlen - EDGE - 1;
    const int base = c - EDGE;

    // ---- 17-tap softmax with max subtraction (matches jax.nn.softmax) ----
    float sc[WIN];
    float m = -3.402823466e38f;
#pragma unroll
    for (int w = 0; w < WIN; ++w) {
        float d = x - s_eval[base + w];
        float s = -(d * d) * tk;
        sc[w] = s;
        m = fmaxf(m, s);
    }

    float esum = 0.0f, vsum = 0.0f;
#pragma unroll
    for (int w = 0; w < WIN; ++w) {
        float e = __expf(sc[w] - m);
        esum += e;
        vsum = fmaf(e, s_vec[base + w], vsum);
    }

    out[gid] = vsum / esum;
}

extern "C" void kernel_launch(void* const* d_in, const int* in_sizes, int n_in,
                              void* d_out, int out_size, void* d_ws, size_t ws_size,
                              hipStream_t stream)
{
    const float* inputs = (const float*)d_in[0];   // (N,1) f32
    const float* eval_t = (const float*)d_in[1];   // (T,1) f32, sorted
    const float* tk_t   = (const float*)d_in[2];   // (T,1) f32
    const float* vec_t  = (const float*)d_in[3];   // (T,1) f32 (V==1)
    float* out = (float*)d_out;

    const int n    = in_sizes[0];
    const int tlen = in_sizes[1];

    const int blocks = (n + BLOCK - 1) / BLOCK;
    hwnet_window_softmax<<<blocks, BLOCK, 0, stream>>>(
        inputs, eval_t, tk_t, vec_t, out, n, tlen);
}